// TransformerConvNet_51754355916842
// MI455X (gfx1250) — compile-verified
//
#include <hip/hip_runtime.h>
#include <math.h>
#include <stdint.h>

// ---------------------------------------------------------------------------
// TransformerConv x3 + mean-pool + FC + log_softmax for MI455X (gfx1250).
// Dense GEMMs: V_WMMA_F32_16X16X4_F32 (full fp32), 4 C-tiles per wave with a
// shared A fragment, K-major (pre-transposed) weights so B-fragments are
// single b64 loads, and software-pipelined fragment prefetch so WMMAs are not
// serialized behind s_wait_loadcnt 0. Edge softmax/aggregation: atomics
// (L2-resident, bandwidth bound).
// ---------------------------------------------------------------------------

typedef float v2f __attribute__((ext_vector_type(2)));
typedef float v8f __attribute__((ext_vector_type(8)));

// ---------------- generic fill ----------------
__global__ void k_fill_u32(unsigned int* __restrict__ p, unsigned int v, long n) {
  long i = (long)blockIdx.x * blockDim.x + threadIdx.x;
  if (i < n) p[i] = v;
}

// ---------------- weight transpose: Wt[n*K + k] = W[k*Nc + n] ---------------
__global__ void k_transpose_w(const float* __restrict__ W,
                              float* __restrict__ Wt, int K, int Nc) {
  int i = blockIdx.x * blockDim.x + threadIdx.x;
  if (i >= K * Nc) return;
  int k = i / Nc, n = i % Nc;
  Wt[(long)n * K + k] = W[i];
}

// ---------------- GEMM: C[M,Nc] = A[M,K] * W[K,Nc] + bias  (fp32 WMMA) -----
// One wave computes FOUR adjacent 16x16 C tiles along N, sharing the A-frag.
// Requires M%16==0, Nc%64==0, K%4==0 (true for all shapes here: M=50000,
// Nc in {192,256}, K in {128,192}). Early-return is wave-uniform, so EXEC is
// all-ones at every WMMA (ISA requirement).
//
// A-frag (16x4, ISA 7.12.2): lane L -> row M = L%16;
//   VGPR0 = A[M][k + (L<16?0:2)], VGPR1 = A[M][k + (L<16?1:3)]
// B-frag (4x16) mirrored, fed from K-major Wt: lane L -> col N = L%16;
//   {Wt[N][k+off], Wt[N][k+off+1]} = one contiguous b64 load.
// C (16x16 f32): VGPR r, lane L -> row = r + (L<16?0:8), col = L%16.
template <int K>
__global__ void k_gemm_wmma_f32_x4(const float* __restrict__ A,
                                   const float* __restrict__ Wt,  // [Nc][K]
                                   const float* __restrict__ bias,
                                   float* __restrict__ C,
                                   int M, int Nc) {
  const int lane = threadIdx.x & 31;
  const int wib  = threadIdx.x >> 5;
  const int wpb  = blockDim.x >> 5;
  const int wid  = blockIdx.x * wpb + wib;
  const int grpN = Nc >> 6;                 // groups of 4 N-tiles
  const int tilesM = M >> 4;
  if (wid >= tilesM * grpN) return;         // wave-uniform
  const int tm  = wid / grpN;
  const int tn0 = (wid % grpN) * 4;         // first of 4 N-tiles

  const int half = lane >> 4;               // 0: lanes 0-15, 1: lanes 16-31
  const int l16  = lane & 15;
  const int rowA = tm * 16 + l16;
  const int col0 = tn0 * 16 + l16;          // cols: col0, +16, +32, +48

  v8f c0, c1, c2, c3;
  {
    const float b0 = bias[col0], b1 = bias[col0 + 16];
    const float b2 = bias[col0 + 32], b3 = bias[col0 + 48];
#pragma unroll
    for (int r = 0; r < 8; ++r) { c0[r] = b0; c1[r] = b1; c2[r] = b2; c3[r] = b3; }
  }

  const float* __restrict__ arow = A + (long)rowA * K;
  const v2f* __restrict__ w0 = (const v2f*)(Wt + (long)(col0)      * K);
  const v2f* __restrict__ w1 = (const v2f*)(Wt + (long)(col0 + 16) * K);
  const v2f* __restrict__ w2 = (const v2f*)(Wt + (long)(col0 + 32) * K);
  const v2f* __restrict__ w3 = (const v2f*)(Wt + (long)(col0 + 48) * K);

  const int koff = half * 2;
  // prologue: fragments for k = 0
  v2f a_c  = *(const v2f*)(arow + koff);
  v2f b0_c = w0[half], b1_c = w1[half], b2_c = w2[half], b3_c = w3[half];

#pragma unroll
  for (int k = 0; k < K; k += 4) {
    v2f a_n, b0_n, b1_n, b2_n, b3_n;
    if (k + 4 < K) {                        // static: K is compile-time
      const int kv = ((k + 4) >> 1) + half; // v2f index of next fragment
      a_n  = *(const v2f*)(arow + (k + 4) + koff);
      b0_n = w0[kv]; b1_n = w1[kv]; b2_n = w2[kv]; b3_n = w3[kv];
    } else {
      a_n = a_c; b0_n = b0_c; b1_n = b1_c; b2_n = b2_c; b3_n = b3_c;
    }
    // 8 args: (neg_a, A, neg_b, B, c_mod, C, reuse_a, reuse_b)
    c0 = __builtin_amdgcn_wmma_f32_16x16x4_f32(false, a_c, false, b0_c, (short)0, c0, false, false);
    c1 = __builtin_amdgcn_wmma_f32_16x16x4_f32(false, a_c, false, b1_c, (short)0, c1, false, false);
    c2 = __builtin_amdgcn_wmma_f32_16x16x4_f32(false, a_c, false, b2_c, (short)0, c2, false, false);
    c3 = __builtin_amdgcn_wmma_f32_16x16x4_f32(false, a_c, false, b3_c, (short)0, c3, false, false);
    a_c = a_n; b0_c = b0_n; b1_c = b1_n; b2_c = b2_n; b3_c = b3_n;
  }

  const int row0 = tm * 16 + half * 8;
#pragma unroll
  for (int r = 0; r < 8; ++r) {
    float* crow = C + (long)(row0 + r) * Nc;
    crow[col0]      = c0[r];
    crow[col0 + 16] = c1[r];
    crow[col0 + 32] = c2[r];
    crow[col0 + 48] = c3[r];
  }
}

// ------------- per-edge attention logits + segment max (encoded atomicMax) --
__device__ __forceinline__ unsigned int enc_f32(float f) {
  unsigned int b = __float_as_uint(f);
  return (b & 0x80000000u) ? ~b : (b | 0x80000000u);   // monotone order map
}
__device__ __forceinline__ float dec_f32(unsigned int m) {
  unsigned int b = (m & 0x80000000u) ? (m & 0x7FFFFFFFu) : ~m;
  return __uint_as_float(b);
}

__global__ void k_edge_logits(const float* __restrict__ q,
                              const float* __restrict__ kk,
                              const int* __restrict__ src,
                              const int* __restrict__ dst,
                              float* __restrict__ logits,
                              unsigned int* __restrict__ segmax,
                              long E, int H, int dout, float scale) {
  long i = (long)blockIdx.x * blockDim.x + threadIdx.x;
  if (i >= E * H) return;
  const int e = (int)(i / H), h = (int)(i % H);
  const int s = src[e], d = dst[e];
  const int dh = H * dout;
  const float* __restrict__ qr = q  + (long)d * dh + h * dout;
  const float* __restrict__ kr = kk + (long)s * dh + h * dout;
  float acc = 0.f;
  for (int t = 0; t < dout; t += 4) {
    float4 qa = *(const float4*)(qr + t);
    float4 ka = *(const float4*)(kr + t);
    acc += qa.x * ka.x + qa.y * ka.y + qa.z * ka.z + qa.w * ka.w;
  }
  acc *= scale;
  logits[i] = acc;
  atomicMax(&segmax[(long)d * H + h], enc_f32(acc));
}

// ------------- exp(logit - segmax[dst]) + segment sum -----------------------
__global__ void k_edge_exp(const float* __restrict__ logits,
                           const unsigned int* __restrict__ segmax,
                           const int* __restrict__ dst,
                           float* __restrict__ ex,
                           float* __restrict__ denom,
                           long E, int H) {
  long i = (long)blockIdx.x * blockDim.x + threadIdx.x;
  if (i >= E * H) return;
  const int e = (int)(i / H), h = (int)(i % H);
  const int d = dst[e];
  const float mx = dec_f32(segmax[(long)d * H + h]);
  const float v = expf(logits[i] - mx);
  ex[i] = v;
  atomicAdd(&denom[(long)d * H + h], v);
}

// ------------- scatter alpha * v[src] into out[dst] (out pre-filled x@Ws+bs)
__global__ void k_edge_scatter(const float* __restrict__ vv,
                               const float* __restrict__ ex,
                               const float* __restrict__ denom,
                               const int* __restrict__ src,
                               const int* __restrict__ dst,
                               float* __restrict__ out,
                               long E, int H, int dout) {
  const int dh = H * dout;
  const int chunks = dh >> 2;
  long i = (long)blockIdx.x * blockDim.x + threadIdx.x;
  if (i >= E * chunks) return;
  const int e = (int)(i / chunks), c = (int)(i % chunks);
  const int d0 = c * 4;
  const int h = d0 / dout;
  const int s = src[e], d = dst[e];
  const float a = ex[(long)e * H + h] / denom[(long)d * H + h];
  float4 vr = *(const float4*)(vv + (long)s * dh + d0);
  float* o = out + (long)d * dh + d0;
  atomicAdd(o + 0, vr.x * a);
  atomicAdd(o + 1, vr.y * a);
  atomicAdd(o + 2, vr.z * a);
  atomicAdd(o + 3, vr.w * a);
}

// ------------- global mean pool (sum + count) -------------------------------
__global__ void k_pool(const float* __restrict__ x,
                       const int* __restrict__ batch,
                       float* __restrict__ sums,
                       float* __restrict__ cnt,
                       long N, int Dm) {
  const int chunks = Dm >> 2;
  long i = (long)blockIdx.x * blockDim.x + threadIdx.x;
  if (i >= N * chunks) return;
  const int n = (int)(i / chunks), c = (int)(i % chunks);
  const int g = batch[n];
  float4 xv = *(const float4*)(x + (long)n * Dm + c * 4);
  float* s = sums + (long)g * Dm + c * 4;
  atomicAdd(s + 0, xv.x);
  atomicAdd(s + 1, xv.y);
  atomicAdd(s + 2, xv.z);
  atomicAdd(s + 3, xv.w);
  if (c == 0) atomicAdd(&cnt[g], 1.0f);
}

// ------------- head: pooled @ fc_w + fc_b, log_softmax ----------------------
__global__ void k_head(const float* __restrict__ sums,
                       const float* __restrict__ cnt,
                       const float* __restrict__ fcw,
                       const float* __restrict__ fcb,
                       float* __restrict__ out,
                       int G, int Dm, int Od) {
  const int g = blockIdx.x * blockDim.x + threadIdx.x;
  if (g >= G) return;
  const float inv = 1.f / fmaxf(cnt[g], 1.f);
  float acc[10];
  for (int o = 0; o < Od; ++o) acc[o] = fcb[o];
  for (int k = 0; k < Dm; ++k) {
    const float p = sums[(long)g * Dm + k] * inv;
    for (int o = 0; o < Od; ++o) acc[o] += p * fcw[k * Od + o];
  }
  float mx = acc[0];
  for (int o = 1; o < Od; ++o) mx = fmaxf(mx, acc[o]);
  float se = 0.f;
  for (int o = 0; o < Od; ++o) se += expf(acc[o] - mx);
  const float lse = logf(se) + mx;
  for (int o = 0; o < Od; ++o) out[(long)g * Od + o] = acc[o] - lse;
}

// ---------------------------------------------------------------------------
extern "C" void kernel_launch(void* const* d_in, const int* in_sizes, int n_in,
                              void* d_out, int out_size, void* d_ws, size_t ws_size,
                              hipStream_t stream) {
  (void)n_in; (void)out_size; (void)ws_size;
  const int IN_FEAT = 128, G = 64, OUT_DIM = 10, FC_IN = 256;
  const long N = in_sizes[0] / IN_FEAT;
  const long E = in_sizes[1] / 2;

  const float* x0    = (const float*)d_in[0];
  const int*   eidx  = (const int*)d_in[1];
  const int*   batch = (const int*)d_in[2];
  const int*   src   = eidx;        // edge_index[0]
  const int*   dst   = eidx + E;    // edge_index[1]
  const float* fcw   = (const float*)d_in[27];
  const float* fcb   = (const float*)d_in[28];

  struct LCfg { int din, dout, h; };
  const LCfg Ls[3] = { {128, 64, 3}, {192, 192, 1}, {192, 256, 1} };

  // ---- carve workspace (256B-aligned slabs, all re-written every call) ----
  char* ws = (char*)d_ws;
  size_t off = 0;
  auto carve = [&](size_t bytes) -> void* {
    void* p = ws + off;
    off += (bytes + 255) & ~(size_t)255;
    return p;
  };
  float* xa     = (float*)carve((size_t)N * 256 * 4);
  float* xb     = (float*)carve((size_t)N * 256 * 4);
  float* qb     = (float*)carve((size_t)N * 256 * 4);
  float* kb     = (float*)carve((size_t)N * 256 * 4);
  float* vb     = (float*)carve((size_t)N * 256 * 4);
  float* logits = (float*)carve((size_t)E * 3 * 4);
  float* exb    = (float*)carve((size_t)E * 3 * 4);
  unsigned int* segmax = (unsigned int*)carve((size_t)N * 3 * 4);
  float* denom  = (float*)carve((size_t)N * 3 * 4);
  float* sums   = (float*)carve((size_t)G * FC_IN * 4);
  float* cnt    = (float*)carve((size_t)G * 4);
  float* wt     = (float*)carve((size_t)192 * 256 * 4);   // K-major weights

  auto blocks = [](long n, int bs) -> unsigned int {
    return (unsigned int)((n + bs - 1) / bs);
  };

  const float* cur = x0;
  float* nxt = xa;
  for (int l = 0; l < 3; ++l) {
    const int K = Ls[l].din, dout = Ls[l].dout, H = Ls[l].h;
    const int dh = H * dout;

    const float* Wmat[4] = { (const float*)d_in[3 + 8 * l + 0],
                             (const float*)d_in[3 + 8 * l + 1],
                             (const float*)d_in[3 + 8 * l + 2],
                             (const float*)d_in[3 + 8 * l + 3] };
    const float* Bvec[4] = { (const float*)d_in[3 + 8 * l + 4],
                             (const float*)d_in[3 + 8 * l + 5],
                             (const float*)d_in[3 + 8 * l + 6],
                             (const float*)d_in[3 + 8 * l + 7] };
    float* Cout[4] = { qb, kb, vb, nxt };   // Ws-output seeds next-layer x

    const long waves = (N / 16) * (dh / 64);       // 4 N-tiles per wave
    const unsigned int gblk = blocks(waves, 4);    // 128 threads = 4 waves
    const unsigned int tblk = blocks((long)K * dh, 256);

    for (int m = 0; m < 4; ++m) {
      k_transpose_w<<<tblk, 256, 0, stream>>>(Wmat[m], wt, K, dh);
      if (K == 128)
        k_gemm_wmma_f32_x4<128><<<gblk, 128, 0, stream>>>(cur, wt, Bvec[m], Cout[m], (int)N, dh);
      else
        k_gemm_wmma_f32_x4<192><<<gblk, 128, 0, stream>>>(cur, wt, Bvec[m], Cout[m], (int)N, dh);
    }

    // segment softmax scratch init
    k_fill_u32<<<blocks(N * H, 256), 256, 0, stream>>>(segmax, 0u, N * H);
    k_fill_u32<<<blocks(N * H, 256), 256, 0, stream>>>((unsigned int*)denom, 0u, N * H);

    const float scale = 1.0f / sqrtf((float)dout);
    k_edge_logits<<<blocks(E * H, 256), 256, 0, stream>>>(
        qb, kb, src, dst, logits, segmax, E, H, dout, scale);
    k_edge_exp<<<blocks(E * H, 256), 256, 0, stream>>>(
        logits, segmax, dst, exb, denom, E, H);
    k_edge_scatter<<<blocks(E * (dh / 4), 256), 256, 0, stream>>>(
        vb, exb, denom, src, dst, nxt, E, H, dout);

    cur = nxt;
    nxt = (nxt == xa) ? xb : xa;
  }

  // global mean pool + classifier head
  k_fill_u32<<<blocks((long)G * FC_IN, 256), 256, 0, stream>>>((unsigned int*)sums, 0u, (long)G * FC_IN);
  k_fill_u32<<<1, 64, 0, stream>>>((unsigned int*)cnt, 0u, G);
  k_pool<<<blocks(N * (FC_IN / 4), 256), 256, 0, stream>>>(cur, batch, sums, cnt, N, FC_IN);
  k_head<<<1, 64, 0, stream>>>(sums, cnt, fcw, fcb, (float*)d_out, G, FC_IN, OUT_DIM);
}